// WlossLayer_69801808495225
// MI455X (gfx1250) — compile-verified
//
#include <hip/hip_runtime.h>

#define EPSF      1e-8f
#define LAMF      100.0f
#define N_ITER    50
#define NA        128
#define BATCH     65536
#define KS        130           // padded LDS row stride (dwords): even (8B align) and bank-conflict-free
#define WAVES     8
#define THREADS   256
#define ROWS_WG   128           // 8 waves * 16 rows

typedef float v2f __attribute__((ext_vector_type(2)));
typedef float v8f __attribute__((ext_vector_type(8)));

#if defined(__HIP_DEVICE_COMPILE__) && !__has_builtin(__builtin_amdgcn_wmma_f32_16x16x4_f32)
#error "need __builtin_amdgcn_wmma_f32_16x16x4_f32 on gfx1250"
#endif

// D(16x16 f32) = A(16x4 f32) * B(4x16 f32) + C
__device__ __forceinline__ v8f wmma_k4(v2f a, v2f b, v8f c) {
#if defined(__HIP_DEVICE_COMPILE__)
  return __builtin_amdgcn_wmma_f32_16x16x4_f32(false, a, false, b, (short)0, c,
                                               false, false);
#else
  (void)a; (void)b;
  return c;  // host stub, never executed
#endif
}

__device__ __forceinline__ float rmax16(float m) {
  // max over each 16-lane half (rows M=r in lanes 0-15, M=r+8 in lanes 16-31)
  m = fmaxf(m, __shfl_xor(m, 1));
  m = fmaxf(m, __shfl_xor(m, 2));
  m = fmaxf(m, __shfl_xor(m, 4));
  m = fmaxf(m, __shfl_xor(m, 8));
  return m;
}
__device__ __forceinline__ float rsum16(float s) {
  s += __shfl_xor(s, 1);
  s += __shfl_xor(s, 2);
  s += __shfl_xor(s, 4);
  s += __shfl_xor(s, 8);
  return s;
}

// acc(16x128, C layout) = A(16x128 from opw) @ Mt^T, where Mt is stored
// row-major-transposed: B-frag lane n gets {Mt[16j+n][k0], Mt[16j+n][k0+1]}
// as one contiguous 8B LDS load.  (If Mt holds X^T, this computes A @ X.)
__device__ __forceinline__ void gemm16_tr(const float* __restrict__ opw,
                                          const float* __restrict__ Mt,
                                          int ln, int half, v8f acc[8]) {
#pragma unroll
  for (int j = 0; j < 8; ++j)
#pragma unroll
    for (int r = 0; r < 8; ++r) acc[j][r] = 0.0f;
  for (int f = 0; f < 32; ++f) {
    int k0 = 4 * f + 2 * half;
    v2f a = *(const v2f*)&opw[ln * KS + k0];
    v2f b[8];
#pragma unroll
    for (int j = 0; j < 8; ++j)
      b[j] = *(const v2f*)&Mt[(16 * j + ln) * KS + k0];
#pragma unroll
    for (int j = 0; j < 8; ++j) acc[j] = wmma_k4(a, b[j], acc[j]);
  }
}

__global__ void zero_out_kernel(float* out) {
  if (threadIdx.x == 0) out[0] = 0.0f;
}

__global__ __launch_bounds__(THREADS)
void sinkhorn_kernel(const float* __restrict__ pred,
                     const float* __restrict__ target,
                     const float* __restrict__ cost,
                     float* __restrict__ out) {
  __shared__ float sK[128 * KS];            // K = exp(-cost/lam), row-major (later: KM)
  __shared__ float sKT[128 * KS];           // K^T, row-major
  __shared__ float sOp[WAVES * 16 * KS];    // per-wave A-operand staging (16x128)

  const int tid  = threadIdx.x;
  const int lane = tid & 31;
  const int wave = tid >> 5;
  const int half = lane >> 4;               // 0: M=r rows / K lo pair, 1: M=r+8 / K hi pair
  const int ln   = lane & 15;
  const int rowBase = blockIdx.x * ROWS_WG + wave * 16;

  // ---- K = exp(-cost/lam) and K^T into LDS ----
  for (int i = tid; i < 128 * 128; i += THREADS) {
    int r = i >> 7, c = i & 127;
    float k = __expf(-cost[i] * (1.0f / LAMF));
    sK[r * KS + c]  = k;
    sKT[c * KS + r] = k;
  }
  __syncthreads();

  float* opw = &sOp[wave * 16 * KS];

  // ---- persistent per-wave state in C-layout fragments ----
  // element (tile j, vgpr r, lane): M = r + 8*half, N = 16*j + ln
  v8f lu[8], lv[8], la[8], lb[8];
  const float LOGU0 = -4.852030263919617f;  // -log(128)
#pragma unroll
  for (int j = 0; j < 8; ++j) {
#pragma unroll
    for (int r = 0; r < 8; ++r) {
      int R = rowBase + r + 8 * half;
      int C = 16 * j + ln;
      la[j][r] = __logf(pred[R * NA + C] + EPSF);
      lb[j][r] = __logf(target[R * NA + C] + EPSF);
      lu[j][r] = LOGU0;
      lv[j][r] = LOGU0;
    }
  }

  float mu[8], mv[8];
  v8f acc[8];

  for (int it = 0; it < N_ITER; ++it) {
    // ---------- u side ----------
#pragma unroll
    for (int r = 0; r < 8; ++r) {
      float m = lu[0][r];
#pragma unroll
      for (int j = 1; j < 8; ++j) m = fmaxf(m, lu[j][r]);
      mu[r] = rmax16(m);
    }
    // u_stab = exp(log_u - max + EPS) -> wave-private LDS (canonical 16x128)
#pragma unroll
    for (int j = 0; j < 8; ++j)
#pragma unroll
      for (int r = 0; r < 8; ++r)
        opw[(r + 8 * half) * KS + 16 * j + ln] = __expf(lu[j][r] - mu[r] + EPSF);

    // S1 = u_stab @ K  (B-frags from K^T copy -> contiguous b64 loads)
    gemm16_tr(opw, sKT, ln, half, acc);

    // log_v = log_b - log(S1) - log_u_max
#pragma unroll
    for (int j = 0; j < 8; ++j)
#pragma unroll
      for (int r = 0; r < 8; ++r)
        lv[j][r] = lb[j][r] - __logf(acc[j][r]) - mu[r];

    // ---------- v side ----------
#pragma unroll
    for (int r = 0; r < 8; ++r) {
      float m = lv[0][r];
#pragma unroll
      for (int j = 1; j < 8; ++j) m = fmaxf(m, lv[j][r]);
      mv[r] = rmax16(m);
    }
#pragma unroll
    for (int j = 0; j < 8; ++j)
#pragma unroll
      for (int r = 0; r < 8; ++r)
        opw[(r + 8 * half) * KS + 16 * j + ln] = __expf(lv[j][r] - mv[r]);

    // S2 = v_stab @ K^T  (B-frags from row-major K)
    gemm16_tr(opw, sK, ln, half, acc);

    // log_u = log_a - log(S2 + EPS) - log_v_max
#pragma unroll
    for (int j = 0; j < 8; ++j)
#pragma unroll
      for (int r = 0; r < 8; ++r)
        lu[j][r] = la[j][r] - __logf(acc[j][r] + EPSF) - mv[r];
  }

  // ---------- epilogue ----------
  // final v_stab from final log_v
#pragma unroll
  for (int r = 0; r < 8; ++r) {
    float m = lv[0][r];
#pragma unroll
    for (int j = 1; j < 8; ++j) m = fmaxf(m, lv[j][r]);
    mv[r] = rmax16(m);
  }
#pragma unroll
  for (int j = 0; j < 8; ++j)
#pragma unroll
    for (int r = 0; r < 8; ++r)
      opw[(r + 8 * half) * KS + 16 * j + ln] = __expf(lv[j][r] - mv[r]);

  __syncthreads();
  // sK := KM = cost * K  (in place; row-major, as needed for @ KM^T access)
  for (int i = tid; i < 128 * 128; i += THREADS) {
    int r = i >> 7, c = i & 127;
    sK[r * KS + c] *= cost[i];
  }
  __syncthreads();

  // S3 = v_stab @ KM^T  (B-frags from row-major KM)
  gemm16_tr(opw, sK, ln, half, acc);

  // wnorm partial: sum exp(log_u + log(S3+EPS) + log_v_max)
  float wpart = 0.0f;
#pragma unroll
  for (int j = 0; j < 8; ++j)
#pragma unroll
    for (int r = 0; r < 8; ++r) {
      float lcp = __logf(acc[j][r] + EPSF) + mv[r];
      wpart += __expf(lu[j][r] + lcp);
    }
  wpart += __shfl_xor(wpart, 1);
  wpart += __shfl_xor(wpart, 2);
  wpart += __shfl_xor(wpart, 4);
  wpart += __shfl_xor(wpart, 8);
  wpart += __shfl_xor(wpart, 16);
  if (lane == 0) atomicAdd(out, wpart * (1.0f / (float)BATCH));

  // grad = center(center(log_u * lam)) / B   (reuse lv as grad storage)
#pragma unroll
  for (int r = 0; r < 8; ++r) {
    float s = 0.0f;
#pragma unroll
    for (int j = 0; j < 8; ++j) { lv[j][r] = lu[j][r] * LAMF; s += lv[j][r]; }
    float mean1 = rsum16(s) * (1.0f / 128.0f);
    float s2 = 0.0f;
#pragma unroll
    for (int j = 0; j < 8; ++j) { lv[j][r] -= mean1; s2 += lv[j][r]; }
    float mean2 = rsum16(s2) * (1.0f / 128.0f);
#pragma unroll
    for (int j = 0; j < 8; ++j)
      lv[j][r] = (lv[j][r] - mean2) * (1.0f / (float)BATCH);
  }
  float* grad = out + 1;
#pragma unroll
  for (int j = 0; j < 8; ++j)
#pragma unroll
    for (int r = 0; r < 8; ++r)
      grad[(rowBase + r + 8 * half) * NA + 16 * j + ln] = lv[j][r];
}

extern "C" void kernel_launch(void* const* d_in, const int* in_sizes, int n_in,
                              void* d_out, int out_size, void* d_ws, size_t ws_size,
                              hipStream_t stream) {
  const float* pred   = (const float*)d_in[0];
  const float* target = (const float*)d_in[1];
  const float* cost   = (const float*)d_in[2];
  float* out = (float*)d_out;
  (void)in_sizes; (void)n_in; (void)out_size; (void)d_ws; (void)ws_size;

  zero_out_kernel<<<1, 32, 0, stream>>>(out);
  sinkhorn_kernel<<<BATCH / ROWS_WG, THREADS, 0, stream>>>(pred, target, cost, out);
}